// GCNSpectrum_30442728194131
// MI455X (gfx1250) — compile-verified
//
#include <hip/hip_runtime.h>
#include <hip/hip_bf16.h>
#include <math.h>

typedef __attribute__((ext_vector_type(2))) float v2f;
typedef __attribute__((ext_vector_type(8))) float v8f;

#define N_NODES 50000
#define N_EDGES 1600000
#define N_GRAPHS 512
#define FEAT 128
#define HID 128
#define N_LAYERS 4
#define MAX_MZ 1000

// ---------------------------------------------------------------------------
// degree / norm precompute
// ---------------------------------------------------------------------------
__global__ void degree_kernel(const int* __restrict__ src, const int* __restrict__ dst,
                              int* __restrict__ dego, int* __restrict__ degi, int nE) {
    int e = blockIdx.x * blockDim.x + threadIdx.x;
    if (e < nE) {
        atomicAdd(&dego[src[e]], 1);
        atomicAdd(&degi[dst[e]], 1);
    }
}

__global__ void norm_kernel(const int* __restrict__ dego, const int* __restrict__ degi,
                            float* __restrict__ nout, float* __restrict__ nin, int n) {
    int i = blockIdx.x * blockDim.x + threadIdx.x;
    if (i < n) {
        nout[i] = rsqrtf(fmaxf((float)dego[i], 1.0f));
        nin[i]  = rsqrtf(fmaxf((float)degi[i], 1.0f));
    }
}

// BN eval-mode folded into scale/shift: y = s*x + t
__global__ void bn_prep_kernel(const float* __restrict__ gamma, const float* __restrict__ beta,
                               const float* __restrict__ mean, const float* __restrict__ var,
                               float* __restrict__ s, float* __restrict__ t, int n) {
    int i = blockIdx.x * blockDim.x + threadIdx.x;
    if (i < n) {
        float sv = gamma[i] * rsqrtf(var[i] + 1e-5f);
        s[i] = sv;
        t[i] = beta[i] - mean[i] * sv;
    }
}

// transpose all GCN weights: Wt[L][j][k] = W[L][k][j]  (B-frag loads become b64)
__global__ void transpose_w_kernel(const float* __restrict__ W, float* __restrict__ Wt) {
    int i = blockIdx.x * blockDim.x + threadIdx.x;   // over N_LAYERS*HID*HID
    if (i < N_LAYERS * HID * HID) {
        int L = i >> 14;          // /16384
        int r = i & 16383;
        int k = r >> 7;           // source row
        int j = r & 127;          // source col
        Wt[(L << 14) + j * HID + k] = W[i];
    }
}

// ---------------------------------------------------------------------------
// edge aggregation: agg[dst] += h[src] * norm_out[src]
// one wave (32 lanes) per edge; float4 per lane => 128 feats, coalesced 512B
// buffers are L2-resident (25.6MB << 192MB), atomics resolve in L2
// ---------------------------------------------------------------------------
__global__ void aggregate_kernel(const float* __restrict__ Hin, const float* __restrict__ norm_out,
                                 const int* __restrict__ src, const int* __restrict__ dst,
                                 float* __restrict__ agg, int nE) {
    int tid  = blockIdx.x * blockDim.x + threadIdx.x;
    int e    = tid >> 5;
    int lane = tid & 31;
    if (e >= nE) return;
    int s = src[e];
    int d = dst[e];
    float no = norm_out[s];
    const float4* hp = (const float4*)(Hin + (size_t)s * FEAT);
    float4 v = hp[lane];
    float* ap = agg + (size_t)d * FEAT + lane * 4;
    atomicAdd(ap + 0, v.x * no);
    atomicAdd(ap + 1, v.y * no);
    atomicAdd(ap + 2, v.z * no);
    atomicAdd(ap + 3, v.w * no);
}

// ---------------------------------------------------------------------------
// fused agg @ W -> *norm_in + b -> ReLU -> BN, via V_WMMA_F32_16X16X4_F32
// norm_in row-scaling hoisted out of K-loop: (diag(n)A)W = diag(n)(AW)
// block = 256 thr = 8 waves; block handles 16 rows x 128 cols; wave w = col tile w
// A frag 16x4: lane 0-15 -> M, koff 0; lanes 16-31 -> koff 2; VGPR0=K, VGPR1=K+1
// B frag 4x16: same K striping, N = lane&15 (loaded b64 from transposed W)
// C/D 16x16: vgpr r -> M = r + 8*(lane>>4), N = lane&15
// ---------------------------------------------------------------------------
__global__ void gemm_fused_kernel(const float* __restrict__ A,        // agg [N,128]
                                  const float* __restrict__ norm_in,  // [N]
                                  const float* __restrict__ Wt,       // [128,128] (j,k) transposed
                                  const float* __restrict__ bias,     // [128]
                                  const float* __restrict__ bn_s,
                                  const float* __restrict__ bn_t,
                                  float* __restrict__ Hout, int nRows) {
    int wave = threadIdx.x >> 5;
    int lane = threadIdx.x & 31;
    int rowBase = blockIdx.x * 16;
    int colBase = wave * 16;
    int r    = lane & 15;
    int koff = (lane >> 4) << 1;

    int arow = rowBase + r;
    if (arow >= nRows) arow = nRows - 1;   // clamp (no EXEC divergence)
    const float* Ap = A + (size_t)arow * HID + koff;          // contiguous K pair
    const float* Bp = Wt + (size_t)(colBase + r) * HID + koff; // contiguous K pair

    v8f c = {};
    #pragma unroll
    for (int k = 0; k < HID; k += 4) {
        v2f a = *(const v2f*)(Ap + k);   // b64
        v2f b = *(const v2f*)(Bp + k);   // b64
        c = __builtin_amdgcn_wmma_f32_16x16x4_f32(false, a, false, b,
                                                  (short)0, c, false, false);
    }

    int col = colBase + r;
    float s  = bn_s[col];
    float t  = bn_t[col];
    float bb = bias[col];
    int rHalf = (lane >> 4) << 3;
    #pragma unroll
    for (int rr = 0; rr < 8; ++rr) {
        int orow = rowBase + rHalf + rr;
        if (orow < nRows) {
            float v = norm_in[orow] * c[rr] + bb;   // hoisted row scaling + bias
            v = v > 0.0f ? v : 0.0f;                // ReLU
            Hout[(size_t)orow * HID + col] = s * v + t;  // BN
        }
    }
}

// ---------------------------------------------------------------------------
// readout: segment sum + segment max over graph_ids
// ---------------------------------------------------------------------------
__device__ inline void atomicMaxF(float* addr, float v) {
    if (v >= 0.0f) atomicMax((int*)addr, __float_as_int(v));
    else           atomicMin((unsigned int*)addr, (unsigned int)__float_as_int(v));
}

__global__ void fill_neginf_kernel(float* __restrict__ p, int n) {
    int i = blockIdx.x * blockDim.x + threadIdx.x;
    if (i < n) p[i] = __int_as_float(0xFF800000);  // -inf
}

__global__ void readout_kernel(const float* __restrict__ H, const int* __restrict__ gid,
                               float* __restrict__ gsum, float* __restrict__ gmax, int nNodes) {
    int tid  = blockIdx.x * blockDim.x + threadIdx.x;
    int node = tid >> 7;
    int f    = tid & 127;
    if (node >= nNodes) return;
    int g = gid[node];
    float v = H[(size_t)node * FEAT + f];
    atomicAdd(&gsum[g * FEAT + f], v);
    atomicMaxF(&gmax[g * FEAT + f], v);
}

__global__ void fix_max_kernel(float* __restrict__ gmax, int n) {
    int i = blockIdx.x * blockDim.x + threadIdx.x;
    if (i < n) {
        float v = gmax[i];
        if (!(v > -1e38f)) gmax[i] = 0.0f;   // kills -inf from empty graphs
    }
}

// ---------------------------------------------------------------------------
// MLP head: dense + LayerNorm + ReLU (x2), dense + sigmoid
// ---------------------------------------------------------------------------
__global__ void mlp1_kernel(const float* __restrict__ gsum, const float* __restrict__ gmax,
                            const float* __restrict__ W1, const float* __restrict__ b1,
                            const float* __restrict__ lng, const float* __restrict__ lnb,
                            float* __restrict__ x1) {
    int g = blockIdx.x;
    int j = threadIdx.x;                  // 256
    __shared__ float xin[256];
    __shared__ float red[256];
    xin[j] = (j < 128) ? gsum[g * 128 + j] : gmax[g * 128 + (j - 128)];
    __syncthreads();
    float acc = b1[j];
    #pragma unroll 8
    for (int i = 0; i < 256; ++i) acc += xin[i] * W1[i * 256 + j];
    red[j] = acc; __syncthreads();
    for (int s2 = 128; s2 > 0; s2 >>= 1) { if (j < s2) red[j] += red[j + s2]; __syncthreads(); }
    float mu = red[0] * (1.0f / 256.0f);
    __syncthreads();
    float dv = acc - mu;
    red[j] = dv * dv; __syncthreads();
    for (int s2 = 128; s2 > 0; s2 >>= 1) { if (j < s2) red[j] += red[j + s2]; __syncthreads(); }
    float var = red[0] * (1.0f / 256.0f);
    float y = dv * rsqrtf(var + 1e-5f) * lng[j] + lnb[j];
    x1[g * 256 + j] = y > 0.0f ? y : 0.0f;
}

__global__ void mlp2_kernel(const float* __restrict__ x1,
                            const float* __restrict__ W2, const float* __restrict__ b2,
                            const float* __restrict__ lng, const float* __restrict__ lnb,
                            float* __restrict__ x2) {
    int g = blockIdx.x;
    int j = threadIdx.x;                  // 128
    __shared__ float xin[256];
    __shared__ float red[128];
    xin[j]       = x1[g * 256 + j];
    xin[j + 128] = x1[g * 256 + j + 128];
    __syncthreads();
    float acc = b2[j];
    #pragma unroll 8
    for (int i = 0; i < 256; ++i) acc += xin[i] * W2[i * 128 + j];
    red[j] = acc; __syncthreads();
    for (int s2 = 64; s2 > 0; s2 >>= 1) { if (j < s2) red[j] += red[j + s2]; __syncthreads(); }
    float mu = red[0] * (1.0f / 128.0f);
    __syncthreads();
    float dv = acc - mu;
    red[j] = dv * dv; __syncthreads();
    for (int s2 = 64; s2 > 0; s2 >>= 1) { if (j < s2) red[j] += red[j + s2]; __syncthreads(); }
    float var = red[0] * (1.0f / 128.0f);
    float y = dv * rsqrtf(var + 1e-5f) * lng[j] + lnb[j];
    x2[g * 128 + j] = y > 0.0f ? y : 0.0f;
}

__global__ void mlp3_kernel(const float* __restrict__ x2,
                            const float* __restrict__ W3, const float* __restrict__ b3,
                            float* __restrict__ out) {
    int g = blockIdx.x;
    __shared__ float xin[128];
    if (threadIdx.x < 128) xin[threadIdx.x] = x2[g * 128 + threadIdx.x];
    __syncthreads();
    for (int j = threadIdx.x; j < MAX_MZ; j += blockDim.x) {
        float acc = b3[j];
        #pragma unroll 8
        for (int i = 0; i < 128; ++i) acc += xin[i] * W3[i * MAX_MZ + j];
        out[(size_t)g * MAX_MZ + j] = 1.0f / (1.0f + __expf(-acc));
    }
}

// ---------------------------------------------------------------------------
extern "C" void kernel_launch(void* const* d_in, const int* in_sizes, int n_in,
                              void* d_out, int out_size, void* d_ws, size_t ws_size,
                              hipStream_t stream) {
    const float* nodef = (const float*)d_in[0];
    const int*   eidx  = (const int*)d_in[1];
    const int*   src   = eidx;
    const int*   dst   = eidx + N_EDGES;
    const int*   gid   = (const int*)d_in[2];
    const float* gcnW  = (const float*)d_in[3];
    const float* gcnb  = (const float*)d_in[4];
    const float* bng   = (const float*)d_in[5];
    const float* bnb   = (const float*)d_in[6];
    const float* bnm   = (const float*)d_in[7];
    const float* bnv   = (const float*)d_in[8];
    const float* W1    = (const float*)d_in[9];
    const float* b1    = (const float*)d_in[10];
    const float* ln1g  = (const float*)d_in[11];
    const float* ln1b  = (const float*)d_in[12];
    const float* W2    = (const float*)d_in[13];
    const float* b2    = (const float*)d_in[14];
    const float* ln2g  = (const float*)d_in[15];
    const float* ln2b  = (const float*)d_in[16];
    const float* W3    = (const float*)d_in[17];
    const float* b3    = (const float*)d_in[18];
    float* out = (float*)d_out;

    // workspace carve (floats)
    float* wsf = (float*)d_ws;
    size_t o = 0;
    float* h0   = wsf + o; o += (size_t)N_NODES * HID;   // agg buffer
    float* h1   = wsf + o; o += (size_t)N_NODES * HID;   // node state
    float* nout = wsf + o; o += N_NODES;
    float* nin  = wsf + o; o += N_NODES;
    int*   dego = (int*)(wsf + o); o += N_NODES;
    int*   degi = (int*)(wsf + o); o += N_NODES;
    float* bns  = wsf + o; o += N_LAYERS * HID;
    float* bnt  = wsf + o; o += N_LAYERS * HID;
    float* wt   = wsf + o; o += (size_t)N_LAYERS * HID * HID;  // transposed weights
    float* gsum = wsf + o; o += (size_t)N_GRAPHS * HID;
    float* gmax = wsf + o; o += (size_t)N_GRAPHS * HID;
    float* x1   = wsf + o; o += (size_t)N_GRAPHS * 256;
    float* x2   = wsf + o; o += (size_t)N_GRAPHS * 128;

    // degrees + norms + folded BN params + transposed weights
    hipMemsetAsync(dego, 0, 2 * N_NODES * sizeof(int), stream);
    degree_kernel<<<(N_EDGES + 255) / 256, 256, 0, stream>>>(src, dst, dego, degi, N_EDGES);
    norm_kernel<<<(N_NODES + 255) / 256, 256, 0, stream>>>(dego, degi, nout, nin, N_NODES);
    bn_prep_kernel<<<(N_LAYERS * HID + 255) / 256, 256, 0, stream>>>(bng, bnb, bnm, bnv, bns, bnt, N_LAYERS * HID);
    transpose_w_kernel<<<(N_LAYERS * HID * HID + 255) / 256, 256, 0, stream>>>(gcnW, wt);

    // 4 GCN layers
    const float* Hcur = nodef;
    for (int L = 0; L < N_LAYERS; ++L) {
        hipMemsetAsync(h0, 0, (size_t)N_NODES * HID * sizeof(float), stream);
        aggregate_kernel<<<(N_EDGES * 32) / 256, 256, 0, stream>>>(Hcur, nout, src, dst, h0, N_EDGES);
        gemm_fused_kernel<<<N_NODES / 16, 256, 0, stream>>>(
            h0, nin, wt + (size_t)L * HID * HID, gcnb + L * HID,
            bns + L * HID, bnt + L * HID, h1, N_NODES);
        Hcur = h1;
    }

    // readout
    hipMemsetAsync(gsum, 0, (size_t)N_GRAPHS * HID * sizeof(float), stream);
    fill_neginf_kernel<<<(N_GRAPHS * HID + 255) / 256, 256, 0, stream>>>(gmax, N_GRAPHS * HID);
    readout_kernel<<<((size_t)N_NODES * HID + 255) / 256, 256, 0, stream>>>(h1, gid, gsum, gmax, N_NODES);
    fix_max_kernel<<<(N_GRAPHS * HID + 255) / 256, 256, 0, stream>>>(gmax, N_GRAPHS * HID);

    // MLP head
    mlp1_kernel<<<N_GRAPHS, 256, 0, stream>>>(gsum, gmax, W1, b1, ln1g, ln1b, x1);
    mlp2_kernel<<<N_GRAPHS, 128, 0, stream>>>(x1, W2, b2, ln2g, ln2b, x2);
    mlp3_kernel<<<N_GRAPHS, 256, 0, stream>>>(x2, W3, b3, out);
}